// RouteEmbeddingMPNN_12549894439068
// MI455X (gfx1250) — compile-verified
//
#include <hip/hip_runtime.h>
#include <hip/hip_bf16.h>

typedef __attribute__((ext_vector_type(2))) float v2f;
typedef __attribute__((ext_vector_type(8))) float v8f;

#define N_NODES 100000
#define N_EDGES 600000
#define CH      128

// ---------------------------------------------------------------------------
// Fill a float buffer with a constant (used to zero accumulators / degree).
// ---------------------------------------------------------------------------
__global__ __launch_bounds__(256)
void fill_f32(float* __restrict__ p, long long n, float val) {
    long long i = (long long)blockIdx.x * 256 + threadIdx.x;
    if (i < n) p[i] = val;
}

// ---------------------------------------------------------------------------
// deg[row[e]] += 1.0f  (fp32 atomic, matches reference segment_sum of ones)
// ---------------------------------------------------------------------------
__global__ __launch_bounds__(256)
void deg_accum(const int* __restrict__ row, float* __restrict__ deg, int nedges) {
    int e = blockIdx.x * 256 + threadIdx.x;
    if (e < nedges) unsafeAtomicAdd(&deg[row[e]], 1.0f);
}

// deg[i] = 1 / max(deg[i], 1)
__global__ __launch_bounds__(256)
void deg_finalize(float* __restrict__ deg, int n) {
    int i = blockIdx.x * 256 + threadIdx.x;
    if (i < n) deg[i] = 1.0f / fmaxf(deg[i], 1.0f);
}

// ---------------------------------------------------------------------------
// h0 = x @ W + b using V_WMMA_F32_16X16X4_F32 (fp32-exact matrix path).
// Block = 256 threads = 8 waves; wave w computes the 16x16 tile at
// rows [16*blockIdx.x, +16), cols [16*w, +16). K=128 -> 32 WMMA steps.
//
// Fragment layouts (ISA 7.12.2, wave32):
//  A 16x4 f32 : lane l holds row m=l&15, K values {2h, 2h+1}, h=l>>4 (2 VGPRs)
//  B 4x16 f32 : lane l holds col n=l&15, K rows  {2h, 2h+1}            (2 VGPRs)
//  C/D 16x16  : lane l, VGPR v -> element (m = v + 8h, n = l&15)        (8 VGPRs)
// ---------------------------------------------------------------------------
__global__ __launch_bounds__(256)
void gemm_bias_wmma(const float* __restrict__ x, const float* __restrict__ W,
                    const float* __restrict__ bias, float* __restrict__ out) {
    const int wave = threadIdx.x >> 5;   // 0..7 -> column tile
    const int lane = threadIdx.x & 31;
    const int m = lane & 15;
    const int h = lane >> 4;             // 0 or 1
    const int row0 = blockIdx.x * 16;
    const int n = wave * 16 + m;         // this lane's output column

    const float* __restrict__ arow = x + (size_t)(row0 + m) * CH;

    v8f c = {};
#pragma unroll 4
    for (int k = 0; k < CH; k += 4) {
        const int ka = k + 2 * h;
        v2f a;
        a.x = arow[ka];
        a.y = arow[ka + 1];
        v2f bfrag;
        bfrag.x = W[(size_t)ka * CH + n];
        bfrag.y = W[(size_t)(ka + 1) * CH + n];
        c = __builtin_amdgcn_wmma_f32_16x16x4_f32(
                /*neg_a=*/false, a, /*neg_b=*/false, bfrag,
                /*c_mod=*/(short)0, c, /*reuse_a=*/false, /*reuse_b=*/false);
    }

    const float bv = bias[n];
#pragma unroll
    for (int v = 0; v < 8; ++v) {
        out[(size_t)(row0 + v + 8 * h) * CH + n] = c[v] + bv;
    }
}

// ---------------------------------------------------------------------------
// One aggregation hop (accumulate only, scaling fused into the gather):
//   acc[row[e]][c] += h_in[col[e]][c] * (deg_inv ? deg_inv[col[e]] : 1)
// One wave (32 lanes) per edge; each lane moves 4 channels (float4 gather,
// 4x global_atomic_add_f32 no-return). Both buffers are 51.2 MB -> L2 resident.
// ---------------------------------------------------------------------------
__global__ __launch_bounds__(256)
void scatter_hop(const float* __restrict__ h_in, const int* __restrict__ row,
                 const int* __restrict__ col, const float* __restrict__ deg_inv,
                 float* __restrict__ acc, int nedges) {
    int e = blockIdx.x * 8 + (threadIdx.x >> 5);
    if (e >= nedges) return;
    const int lane = threadIdx.x & 31;
    const int src = col[e];
    const int dst = row[e];

    const float4* pin = (const float4*)(h_in + (size_t)src * CH) + lane;
    float4 v = *pin;
    const float s = deg_inv ? deg_inv[src] : 1.0f;
    v.x *= s; v.y *= s; v.z *= s; v.w *= s;

    float* pout = acc + (size_t)dst * CH + lane * 4;
    unsafeAtomicAdd(pout + 0, v.x);
    unsafeAtomicAdd(pout + 1, v.y);
    unsafeAtomicAdd(pout + 2, v.z);
    unsafeAtomicAdd(pout + 3, v.w);
}

// Final elementwise scale: out[i][c] *= deg_inv[i]
__global__ __launch_bounds__(256)
void scale_rows(float* __restrict__ out, const float* __restrict__ deg_inv, int n) {
    int i = blockIdx.x * 256 + threadIdx.x;
    if (i < n) out[i] *= deg_inv[i >> 7];   // CH == 128
}

// ---------------------------------------------------------------------------
extern "C" void kernel_launch(void* const* d_in, const int* in_sizes, int n_in,
                              void* d_out, int out_size, void* d_ws, size_t ws_size,
                              hipStream_t stream) {
    const float* x    = (const float*)d_in[0];   // [N, 128]
    const int*   edge = (const int*)d_in[1];     // [2, E]
    const float* W    = (const float*)d_in[2];   // [128, 128]
    const float* b    = (const float*)d_in[3];   // [128]
    float* out = (float*)d_out;                  // [N, 128]

    const int N = N_NODES, E = N_EDGES;
    const long long NC = (long long)N * CH;      // 12.8M floats

    const int* row = edge;         // edge_index[0]
    const int* col = edge + E;     // edge_index[1]

    // Workspace layout: [deg_inv : N floats, padded] [hbuf : N*CH floats]
    float* deg  = (float*)d_ws;
    float* hbuf = (float*)d_ws + ((N + 1023) & ~1023);

    dim3 blk(256);
    const int gN   = (N + 255) / 256;
    const int gE   = (E + 255) / 256;
    const int gNC  = (int)((NC + 255) / 256);
    const int gEw  = (E + 7) / 8;      // 8 edges (waves) per block
    const int gMM  = N / 16;           // 100000 / 16 = 6250 row tiles

    // Degree -> deg_inv
    fill_f32<<<gN, blk, 0, stream>>>(deg, N, 0.0f);
    deg_accum<<<gE, blk, 0, stream>>>(row, deg, E);
    deg_finalize<<<gN, blk, 0, stream>>>(deg, N);

    // h0 = x @ W + b  (fp32 WMMA)
    gemm_bias_wmma<<<gMM, blk, 0, stream>>>(x, W, b, hbuf);

    // Hop 1: acc1(out) = sum h0[col]
    fill_f32<<<gNC, blk, 0, stream>>>(out, NC, 0.0f);
    scatter_hop<<<gEw, blk, 0, stream>>>(hbuf, row, col, (const float*)nullptr, out, E);

    // Hop 2: acc2(hbuf) = sum acc1[col] * deg_inv[col]
    fill_f32<<<gNC, blk, 0, stream>>>(hbuf, NC, 0.0f);
    scatter_hop<<<gEw, blk, 0, stream>>>(out, row, col, deg, hbuf, E);

    // Hop 3: acc3(out) = sum acc2[col] * deg_inv[col]
    fill_f32<<<gNC, blk, 0, stream>>>(out, NC, 0.0f);
    scatter_hop<<<gEw, blk, 0, stream>>>(hbuf, row, col, deg, out, E);

    // Final mean scaling of hop 3
    scale_rows<<<gNC, blk, 0, stream>>>(out, deg, (int)NC);
}